// Regulator_32693291057706
// MI455X (gfx1250) — compile-verified
//
#include <hip/hip_runtime.h>
#include <hip/hip_bf16.h>

#define B_    16
#define T_    1024
#define S_    256
#define C_    256
#define EPS_  1e-3f

typedef float v2f __attribute__((ext_vector_type(2)));
typedef float v8f __attribute__((ext_vector_type(8)));

__device__ __forceinline__ v8f wmma4(v2f a, v2f b, v8f c) {
    return __builtin_amdgcn_wmma_f32_16x16x4_f32(false, a, false, b, (short)0, c,
                                                 false, false);
}

// fast reciprocal: single v_rcp_f32, no IEEE div fixup chain
__device__ __forceinline__ float frcp(float x) {
    return __builtin_amdgcn_rcpf(x);
}

__device__ __forceinline__ float silu(float x) {
    return x * frcp(1.f + __expf(-x));
}

// ---------------------------------------------------------------- cumsum scan
__global__ __launch_bounds__(256)
void k_scan(const float* __restrict__ dur, float* __restrict__ start0,
            float* __restrict__ end0) {
    __shared__ float buf[S_];
    int b = blockIdx.x, s = threadIdx.x;
    float d = dur[b * S_ + s];
    buf[s] = d;
    __syncthreads();
    for (int off = 1; off < S_; off <<= 1) {
        float v = (s >= off) ? buf[s - off] : 0.f;
        __syncthreads();
        buf[s] += v;
        __syncthreads();
    }
    float e = buf[s];
    end0[b * S_ + s]   = e;
    start0[b * S_ + s] = e - d;
}

// ------------------------------------------------- conv(K=3,C->8)+LN+SiLU x2
__global__ __launch_bounds__(256)
void k_conv(const float* __restrict__ ctx,
            const float* __restrict__ wcw, const float* __restrict__ bcw,
            const float* __restrict__ gw,  const float* __restrict__ bw,
            const float* __restrict__ wca, const float* __restrict__ bca,
            const float* __restrict__ ga,  const float* __restrict__ ba,
            float* __restrict__ hW, float* __restrict__ hA) {
    __shared__ float red[256][17];
    int bs = blockIdx.x;
    int b  = bs / S_, s = bs % S_;
    int c  = threadIdx.x;

    float acc[16];
#pragma unroll
    for (int o = 0; o < 16; o++) acc[o] = 0.f;

    for (int k = 0; k < 3; k++) {
        int ss = s + k - 1;
        if (ss < 0 || ss >= S_) continue;
        float x = ctx[((size_t)b * S_ + ss) * C_ + c];
        const float* wk = wcw + (k * C_ + c) * 8;   // w[k][c][o]
        const float* ak = wca + (k * C_ + c) * 8;
#pragma unroll
        for (int o = 0; o < 8; o++) {
            acc[o]     += x * wk[o];
            acc[8 + o] += x * ak[o];
        }
    }
#pragma unroll
    for (int o = 0; o < 16; o++) red[c][o] = acc[o];
    __syncthreads();
    for (int st = 128; st >= 1; st >>= 1) {
        if (c < st)
            for (int o = 0; o < 16; o++) red[c][o] += red[c + st][o];
        __syncthreads();
    }
    if (c < 16) {
        int grp = c >> 3, o = c & 7;
        const float* cb = grp ? bca : bcw;
        float h[8], mu = 0.f;
#pragma unroll
        for (int i = 0; i < 8; i++) { h[i] = red[0][grp * 8 + i] + cb[i]; mu += h[i]; }
        mu *= 0.125f;
        float var = 0.f;
#pragma unroll
        for (int i = 0; i < 8; i++) { float d = h[i] - mu; var += d * d; }
        var *= 0.125f;
        float inv = rsqrtf(var + EPS_);
        float g  = (grp ? ga : gw)[o];
        float bb = (grp ? ba : bw)[o];
        float v  = silu((h[o] - mu) * inv * g + bb);
        (grp ? hA : hW)[((size_t)b * S_ + s) * 8 + o] = v;
    }
}

// ------------------------------------------------- per-wave WMMA MLP 10->16->16
// Returns post-SiLU layer-2 output in C-layout (row M = r + 8*hi, col N = n).
__device__ __forceinline__ v8f mlp16(const float* __restrict__ hrow, float f8, float f9,
                                     const float (&B1)[3][2], float bias1,
                                     const float (&B2)[4][2], float bias2,
                                     float* __restrict__ twave, int n, int hi) {
    // A-operand values for the 3 layer-1 K-steps
    float xv[3][2];
#pragma unroll
    for (int kap = 0; kap < 3; kap++)
#pragma unroll
        for (int j = 0; j < 2; j++) {
            int kk = kap * 4 + 2 * hi + j;
            float v;
            if (kk < 8)       v = hrow[kk];
            else if (kk == 8) v = f8;
            else if (kk == 9) v = f9;
            else              v = 0.f;
            xv[kap][j] = v;
        }
    v8f c1 = {0.f, 0.f, 0.f, 0.f, 0.f, 0.f, 0.f, 0.f};
#pragma unroll
    for (int kap = 0; kap < 3; kap++) {
        v2f a;  a.x = xv[kap][0]; a.y = xv[kap][1];
        v2f bb; bb.x = B1[kap][0]; bb.y = B1[kap][1];
        c1 = wmma4(a, bb, c1);
    }
    // bias + SiLU, stash into LDS for the C->A transpose
#pragma unroll
    for (int r = 0; r < 8; r++) {
        float v = silu(c1[r] + bias1);
        twave[(r + 8 * hi) * 17 + n] = v;
    }
    v8f c2 = {0.f, 0.f, 0.f, 0.f, 0.f, 0.f, 0.f, 0.f};
#pragma unroll
    for (int kap = 0; kap < 4; kap++) {
        v2f a;
        a.x = twave[n * 17 + kap * 4 + 2 * hi + 0];
        a.y = twave[n * 17 + kap * 4 + 2 * hi + 1];
        v2f bb; bb.x = B2[kap][0]; bb.y = B2[kap][1];
        c2 = wmma4(a, bb, c2);
    }
#pragma unroll
    for (int r = 0; r < 8; r++) c2[r] = silu(c2[r] + bias2);
    return c2;
}

// ---------------------------------- logits -> softmax -> w (out) and aux (ws)
__global__ __launch_bounds__(256)
void k_blend(const float* __restrict__ hW, const float* __restrict__ hA,
             const float* __restrict__ start0, const float* __restrict__ end0,
             const float* __restrict__ mask,
             const float* __restrict__ w1w, const float* __restrict__ b1w,
             const float* __restrict__ w2w, const float* __restrict__ b2w,
             const float* __restrict__ w_pw, const float* __restrict__ b_pw,
             const float* __restrict__ w1a, const float* __restrict__ b1a,
             const float* __restrict__ w2a, const float* __restrict__ b2a,
             float* __restrict__ w_out, float* __restrict__ aux_out) {
    const int bt = blockIdx.x;
    const int b = bt / T_, t = bt % T_;
    const int tid = threadIdx.x, wave = tid >> 5, lane = tid & 31;
    const int n = lane & 15, hi = lane >> 4;

    __shared__ float s_logit[S_];
    __shared__ float s_w[S_];
    __shared__ float s_red[S_];
    __shared__ float s_auxp[8 * 16];
    __shared__ float tbuf[8][16 * 17];

    // per-lane weight registers (B-operand layout: K = kap*4 + 2*hi + j, N = n)
    float B1w[3][2], B1a[3][2];
#pragma unroll
    for (int kap = 0; kap < 3; kap++)
#pragma unroll
        for (int j = 0; j < 2; j++) {
            int kk = kap * 4 + 2 * hi + j;
            B1w[kap][j] = (kk < 10) ? w1w[kk * 16 + n] : 0.f;
            B1a[kap][j] = (kk < 10) ? w1a[kk * 16 + n] : 0.f;
        }
    float B2w[4][2], B2a[4][2];
#pragma unroll
    for (int kap = 0; kap < 4; kap++)
#pragma unroll
        for (int j = 0; j < 2; j++) {
            int kk = kap * 4 + 2 * hi + j;
            B2w[kap][j] = w2w[kk * 16 + n];
            B2a[kap][j] = w2a[kk * 16 + n];
        }
    float bias1w = b1w[n], bias2w = b2w[n];
    float bias1a = b1a[n], bias2a = b2a[n];
    float wp = w_pw[n], bp = b_pw[0];

    // ---- logits: blender_w MLP + projection, 2 s-tiles per wave
    for (int ti = 0; ti < 2; ti++) {
        int s0   = (wave * 2 + ti) * 16;
        int srow = s0 + n;
        const float* hrow = hW + ((size_t)b * S_ + srow) * 8;
        float st = start0[b * S_ + srow], en = end0[b * S_ + srow];
        float f8 = (float)t - st;
        float f9 = en + st - (float)t;
        v8f c2 = mlp16(hrow, f8, f9, B1w, bias1w, B2w, bias2w, &tbuf[wave][0], n, hi);
        float p[8];
#pragma unroll
        for (int r = 0; r < 8; r++) p[r] = c2[r] * wp;
#pragma unroll
        for (int m = 1; m < 16; m <<= 1)
#pragma unroll
            for (int r = 0; r < 8; r++) p[r] += __shfl_xor(p[r], m, 32);
        if (n == 0)
#pragma unroll
            for (int r = 0; r < 8; r++) s_logit[s0 + r + 8 * hi] = p[r] + bp;
    }
    __syncthreads();

    // ---- masked softmax over s (block-wide)
    {
        int s = tid;
        float m0 = mask[(size_t)b * T_ * S_ + s];          // mask[b,0,s]
        float lg = m0 * s_logit[s] + (1.f - m0) * (-1000000.f);
        s_red[s] = lg;
        __syncthreads();
        for (int st = 128; st >= 1; st >>= 1) {
            if (s < st) s_red[s] = fmaxf(s_red[s], s_red[s + st]);
            __syncthreads();
        }
        float mx = s_red[0];
        __syncthreads();
        float e = __expf(lg - mx);
        s_red[s] = e;
        __syncthreads();
        for (int st = 128; st >= 1; st >>= 1) {
            if (s < st) s_red[s] += s_red[s + st];
            __syncthreads();
        }
        float rsum = frcp(s_red[0]);
        float wv = e * rsum * mask[((size_t)b * T_ + t) * S_ + s];
        s_w[s] = wv;
        w_out[((size_t)b * T_ + t) * S_ + s] = wv;
    }
    __syncthreads();

    // ---- aux: blender_a MLP, weighted sum over s with w
    float acc = 0.f;
    for (int ti = 0; ti < 2; ti++) {
        int s0   = (wave * 2 + ti) * 16;
        int srow = s0 + n;
        const float* hrow = hA + ((size_t)b * S_ + srow) * 8;
        float st = start0[b * S_ + srow], en = end0[b * S_ + srow];
        v8f c2 = mlp16(hrow, (float)t - st, en + st - (float)t,
                       B1a, bias1a, B2a, bias2a, &tbuf[wave][0], n, hi);
#pragma unroll
        for (int r = 0; r < 8; r++) acc += c2[r] * s_w[s0 + r + 8 * hi];
    }
    acc += __shfl_xor(acc, 16, 32);
    if (hi == 0) s_auxp[wave * 16 + n] = acc;
    __syncthreads();
    if (tid < 16) {
        float sum = 0.f;
#pragma unroll
        for (int wv = 0; wv < 8; wv++) sum += s_auxp[wv * 16 + tid];
        aux_out[((size_t)b * T_ + t) * 16 + tid] = sum;
    }
}

// ----------------------------------------- aligned = w @ context (per batch)
__global__ __launch_bounds__(256)
void k_att(const float* __restrict__ w, const float* __restrict__ ctx,
           float* __restrict__ out) {
    int gw   = blockIdx.x * 8 + (threadIdx.x >> 5);
    int lane = threadIdx.x & 31, n = lane & 15, hi = lane >> 4;
    const int cT = C_ / 16, tT = T_ / 16;
    int b   = gw / (tT * cT);
    int rem = gw % (tT * cT);
    int t0  = (rem / cT) * 16;
    int c0  = (rem % cT) * 16;
    const float* Wb = w   + (size_t)b * T_ * S_;
    const float* Cb = ctx + (size_t)b * S_ * C_;
    v8f acc = {0.f, 0.f, 0.f, 0.f, 0.f, 0.f, 0.f, 0.f};
    for (int k = 0; k < S_; k += 4) {
        v2f a, bb;
        a.x  = Wb[(size_t)(t0 + n) * S_ + k + 2 * hi + 0];
        a.y  = Wb[(size_t)(t0 + n) * S_ + k + 2 * hi + 1];
        bb.x = Cb[(size_t)(k + 2 * hi + 0) * C_ + c0 + n];
        bb.y = Cb[(size_t)(k + 2 * hi + 1) * C_ + c0 + n];
        acc = wmma4(a, bb, acc);
    }
#pragma unroll
    for (int r = 0; r < 8; r++)
        out[((size_t)b * T_ + t0 + r + 8 * hi) * C_ + c0 + n] = acc[r];
}

// ---------------------------- final: [aligned | aux] @ w_pa + b_pa -> (BT,C)
__global__ __launch_bounds__(256)
void k_proj(const float* __restrict__ alg, const float* __restrict__ aux,
            const float* __restrict__ w_pa, const float* __restrict__ b_pa,
            float* __restrict__ out) {
    int gw   = blockIdx.x * 8 + (threadIdx.x >> 5);
    int lane = threadIdx.x & 31, n = lane & 15, hi = lane >> 4;
    const int cT = C_ / 16;
    int r0 = (gw / cT) * 16;          // row over B*T
    int c0 = (gw % cT) * 16;
    v8f acc = {0.f, 0.f, 0.f, 0.f, 0.f, 0.f, 0.f, 0.f};
    for (int k = 0; k < C_; k += 4) { // aligned part of K
        v2f a, bb;
        a.x  = alg[(size_t)(r0 + n) * C_ + k + 2 * hi + 0];
        a.y  = alg[(size_t)(r0 + n) * C_ + k + 2 * hi + 1];
        bb.x = w_pa[(size_t)(k + 2 * hi + 0) * C_ + c0 + n];
        bb.y = w_pa[(size_t)(k + 2 * hi + 1) * C_ + c0 + n];
        acc = wmma4(a, bb, acc);
    }
#pragma unroll
    for (int k = 0; k < 16; k += 4) { // aux part of K
        v2f a, bb;
        a.x  = aux[(size_t)(r0 + n) * 16 + k + 2 * hi + 0];
        a.y  = aux[(size_t)(r0 + n) * 16 + k + 2 * hi + 1];
        bb.x = w_pa[(size_t)(C_ + k + 2 * hi + 0) * C_ + c0 + n];
        bb.y = w_pa[(size_t)(C_ + k + 2 * hi + 1) * C_ + c0 + n];
        acc = wmma4(a, bb, acc);
    }
    float bias = b_pa[c0 + n];
#pragma unroll
    for (int r = 0; r < 8; r++)
        out[(size_t)(r0 + r + 8 * hi) * C_ + c0 + n] = acc[r] + bias;
}

extern "C" void kernel_launch(void* const* d_in, const int* in_sizes, int n_in,
                              void* d_out, int out_size, void* d_ws, size_t ws_size,
                              hipStream_t stream) {
    const float* ctx     = (const float*)d_in[0];
    const float* dur     = (const float*)d_in[1];
    const float* mask    = (const float*)d_in[2];
    const float* w_convw = (const float*)d_in[3];
    const float* b_convw = (const float*)d_in[4];
    const float* g_lnw   = (const float*)d_in[5];
    const float* b_lnw   = (const float*)d_in[6];
    const float* w1w     = (const float*)d_in[7];
    const float* b1w     = (const float*)d_in[8];
    const float* w2w     = (const float*)d_in[9];
    const float* b2w     = (const float*)d_in[10];
    const float* w_pw    = (const float*)d_in[11];
    const float* b_pw    = (const float*)d_in[12];
    const float* w_conva = (const float*)d_in[13];
    const float* b_conva = (const float*)d_in[14];
    const float* g_lna   = (const float*)d_in[15];
    const float* b_lna   = (const float*)d_in[16];
    const float* w1a     = (const float*)d_in[17];
    const float* b1a     = (const float*)d_in[18];
    const float* w2a     = (const float*)d_in[19];
    const float* b2a     = (const float*)d_in[20];
    const float* w_pa    = (const float*)d_in[21];
    const float* b_pa    = (const float*)d_in[22];

    float* ws     = (float*)d_ws;
    float* start0 = ws;                      // B*S
    float* end0   = ws + 4096;               // B*S
    float* hW     = ws + 8192;               // B*S*8
    float* hA     = ws + 40960;               // B*S*8
    float* aux    = ws + 73728;              // B*T*16
    float* alg    = ws + 335872;             // B*T*C

    float* w_out = (float*)d_out;                        // (B,T,S)
    float* out2  = (float*)d_out + (size_t)B_ * T_ * S_; // (B,T,C)

    k_scan<<<B_, 256, 0, stream>>>(dur, start0, end0);
    k_conv<<<B_ * S_, 256, 0, stream>>>(ctx, w_convw, b_convw, g_lnw, b_lnw,
                                        w_conva, b_conva, g_lna, b_lna, hW, hA);
    k_blend<<<B_ * T_, 256, 0, stream>>>(hW, hA, start0, end0, mask,
                                         w1w, b1w, w2w, b2w, w_pw, b_pw,
                                         w1a, b1a, w2a, b2a, w_out, aux);
    k_att<<<(B_ * (T_ / 16) * (C_ / 16)) / 8, 256, 0, stream>>>(w_out, ctx, alg);
    k_proj<<<((B_ * T_ / 16) * (C_ / 16)) / 8, 256, 0, stream>>>(alg, aux, w_pa,
                                                                 b_pa, out2);
}